// Memory_75892072120755
// MI455X (gfx1250) — compile-verified
//
#include <hip/hip_runtime.h>
#include <hip/hip_bf16.h>

// DNC memory-update step for MI455X (gfx1250), wave32.
// Bandwidth-bound: link matrix (B,N,N)=268MB dominates (~570MB total traffic
// -> ~25us at 23.3 TB/s). WMMA F32_16x16x4 fuses the rank-1 ww_i*prec_j outer
// product into the link tile update on the matrix pipe.

typedef __attribute__((ext_vector_type(2))) float v2f;
typedef __attribute__((ext_vector_type(8))) float v8f;

constexpr int B_ = 16;
constexpr int N_ = 2048;
constexpr int W_ = 128;
constexpr int R_ = 4;

// ---------------------------------------------------------------------------
// Kernel 1: per-batch exclusive cumprod of usage -> allocation -> ww,
// per-batch sum of ww, and usage_new. One block (256 thr) per batch.
// ---------------------------------------------------------------------------
__global__ void dnc_prep(const float* __restrict__ usage,
                         const float* __restrict__ rw,     // read_weights (B,R,N)
                         const float* __restrict__ fg,     // free_gates   (B,R,N)
                         const float* __restrict__ ag,     // allocation_gate (B,1,1)
                         const float* __restrict__ wg,     // write_gate      (B,1,1)
                         float* __restrict__ ws_ww,        // (B,N)
                         float* __restrict__ ws_bsum,      // (B)
                         float* __restrict__ out_usage)    // (B,N)
{
    __shared__ float s_chunk[256];
    __shared__ float s_red[256];
    const int b = blockIdx.x;
    const int t = threadIdx.x;
    const int base = t * 8;                 // 256 threads * 8 = 2048 = N
    const float a_g = ag[b];
    const float w_g = wg[b];

    float u[8], el[8];
    float p = 1.0f;
#pragma unroll
    for (int i = 0; i < 8; ++i) {
        u[i] = usage[b * N_ + base + i];
        el[i] = p;                           // exclusive prefix within chunk
        p *= u[i];
    }
    s_chunk[t] = p;
    __syncthreads();
    if (t == 0) {                            // serial exclusive scan of 256 chunk products
        float run = 1.0f;
        for (int k = 0; k < 256; ++k) {
            float tmp = s_chunk[k];
            s_chunk[k] = run;
            run *= tmp;
        }
    }
    __syncthreads();
    const float off = s_chunk[t];

    float lsum = 0.0f;
#pragma unroll
    for (int i = 0; i < 8; ++i) {
        const int n = base + i;
        const float excl  = off * el[i];
        const float alloc = (1.0f - u[i]) * excl;
        // wca == 1 (softmax over singleton head dim), so:
        const float ww = w_g * (a_g * (alloc - 1.0f) + 1.0f);
        ws_ww[b * N_ + n] = ww;
        lsum += ww;
        float ret = 1.0f;
#pragma unroll
        for (int r = 0; r < R_; ++r) {
            const int idx = (b * R_ + r) * N_ + n;
            ret *= (1.0f - fg[idx] * rw[idx]);
        }
        out_usage[b * N_ + n] = (u[i] + ww - u[i] * ww) * ret;
    }

    s_red[t] = lsum;
    __syncthreads();
    for (int s = 128; s > 0; s >>= 1) {
        if (t < s) s_red[t] += s_red[t + s];
        __syncthreads();
    }
    if (t == 0) ws_bsum[b] = s_red[0];
}

// ---------------------------------------------------------------------------
// Kernel 2: precedence_new = (1 - S) * precedence + ww,  S = global sum of ww.
// ---------------------------------------------------------------------------
__global__ void dnc_prec(const float* __restrict__ prec,
                         const float* __restrict__ ws_ww,
                         const float* __restrict__ ws_bsum,
                         float* __restrict__ out_prec)
{
    const int idx = blockIdx.x * blockDim.x + threadIdx.x;  // 0..B*N-1
    float S = 0.0f;
#pragma unroll
    for (int b = 0; b < B_; ++b) S += ws_bsum[b];           // 16 L2-cached loads
    out_prec[idx] = (1.0f - S) * prec[idx] + ws_ww[idx];
}

// ---------------------------------------------------------------------------
// Kernel 3: memory_new = M*(1 - ww_n*e_w) + ww_n*v_w   (float4-vectorized)
// ---------------------------------------------------------------------------
__global__ void dnc_mem(const float4* __restrict__ mem,
                        const float4* __restrict__ evec,   // (B, W/4)
                        const float4* __restrict__ wvec,   // (B, W/4)
                        const float* __restrict__ ws_ww,   // (B,N)
                        float4* __restrict__ out_mem)
{
    const int idx = blockIdx.x * blockDim.x + threadIdx.x;  // over B*N*W/4
    const int W4 = W_ / 4;
    const int w4 = idx % W4;
    const int bn = idx / W4;                                // b*N + n
    const int b  = bn / N_;
    const float ww = ws_ww[bn];
    const float4 m = mem[idx];
    const float4 e = evec[b * W4 + w4];
    const float4 v = wvec[b * W4 + w4];
    float4 o;
    o.x = m.x * (1.0f - ww * e.x) + ww * v.x;
    o.y = m.y * (1.0f - ww * e.y) + ww * v.y;
    o.z = m.z * (1.0f - ww * e.z) + ww * v.z;
    o.w = m.w * (1.0f - ww * e.w) + ww * v.w;
    out_mem[idx] = o;
}

// ---------------------------------------------------------------------------
// Kernel 4: link_new[b,i,j] = (1 - ww_i - ww_j)*L + ww_i*prec_j.
// One 16x16 (i,j) tile per wave32. The elementwise term goes into C via VALU;
// the rank-1 outer product rides V_WMMA_F32_16X16X4_F32 (K=0 slot only).
// C/D layout: VGPR k holds row (k + 8*(lane>=16)), col = lane&15.
// A (16x4): VGPR0 lanes0-15 = K=0 column (ww_i), everything else 0.
// B (4x16): VGPR0 lanes0-15 = K=0 row (prec_j), everything else 0.
// ---------------------------------------------------------------------------
__global__ void dnc_link(const float* __restrict__ link,
                         const float* __restrict__ prec,   // input precedence (B,N)
                         const float* __restrict__ ws_ww,  // (B,N)
                         float* __restrict__ out_link)
{
    const int lane  = threadIdx.x & 31;
    const int wave  = threadIdx.x >> 5;
    const int gwave = blockIdx.x * (blockDim.x >> 5) + wave;

    const int TJ = N_ / 16;                  // 128 tiles per row
    int t = gwave;
    const int tj = t % TJ; t /= TJ;
    const int ti = t % TJ; t /= TJ;
    const int b  = t;

    const int g = lane >> 4;                 // half-wave group: rows 0-7 vs 8-15
    const int m = lane & 15;                 // column within tile

    const size_t base = (size_t)b * N_ * N_ + (size_t)(ti * 16) * N_ + (size_t)(tj * 16);
    const int wwrow_base = b * N_ + ti * 16;
    const int col_idx    = b * N_ + tj * 16 + m;

    const float wwj = ws_ww[col_idx];
    const float pj  = prec[col_idx];

    v2f a;  a.x = (g == 0) ? ws_ww[wwrow_base + m] : 0.0f;  a.y = 0.0f;   // A: K=0 col
    v2f bm; bm.x = (g == 0) ? pj : 0.0f;                    bm.y = 0.0f;  // B: K=0 row

    v8f c;
#pragma unroll
    for (int k = 0; k < 8; ++k) {
        const int row  = k + 8 * g;
        const float wi = ws_ww[wwrow_base + row];
        const float L  = link[base + (size_t)row * N_ + m];
        c[k] = (1.0f - wi - wwj) * L;
    }

    // D = A x B + C  ->  ww_i*prec_j + (1 - ww_i - ww_j)*L
    v8f d = __builtin_amdgcn_wmma_f32_16x16x4_f32(
        /*neg_a=*/false, a, /*neg_b=*/false, bm,
        /*c_mod=*/(short)0, c, /*reuse_a=*/false, /*reuse_b=*/false);

#pragma unroll
    for (int k = 0; k < 8; ++k) {
        const int row = k + 8 * g;
        out_link[base + (size_t)row * N_ + m] = d[k];
    }
}

// ---------------------------------------------------------------------------
extern "C" void kernel_launch(void* const* d_in, const int* in_sizes, int n_in,
                              void* d_out, int out_size, void* d_ws, size_t ws_size,
                              hipStream_t stream) {
    const float* memory = (const float*)d_in[0];   // (B,N,W)
    const float* link   = (const float*)d_in[1];   // (B,N,N)
    const float* usage  = (const float*)d_in[2];   // (B,1,N)
    const float* prec   = (const float*)d_in[3];   // (B,1,N)
    const float* rw     = (const float*)d_in[4];   // (B,R,N)
    // d_in[5] write_key, d_in[6] write_strength: dead (softmax over singleton)
    const float* ag     = (const float*)d_in[7];   // (B,1,1)
    const float* wg     = (const float*)d_in[8];   // (B,1,1)
    const float* wvec   = (const float*)d_in[9];   // (B,1,W)
    const float* evec   = (const float*)d_in[10];  // (B,1,W)
    const float* fg     = (const float*)d_in[11];  // (B,R,N)

    float* out       = (float*)d_out;
    float* out_mem   = out;                                   // B*N*W
    float* out_link  = out_mem + (size_t)B_ * N_ * W_;        // B*N*N
    float* out_prec  = out_link + (size_t)B_ * N_ * N_;       // B*N
    float* out_usage = out_prec + (size_t)B_ * N_;            // B*N

    float* ws_ww   = (float*)d_ws;                            // B*N floats
    float* ws_bsum = ws_ww + (size_t)B_ * N_;                 // B floats

    dnc_prep<<<B_, 256, 0, stream>>>(usage, rw, fg, ag, wg, ws_ww, ws_bsum, out_usage);

    dnc_prec<<<(B_ * N_) / 256, 256, 0, stream>>>(prec, ws_ww, ws_bsum, out_prec);

    dnc_mem<<<(B_ * N_ * W_ / 4) / 256, 256, 0, stream>>>(
        (const float4*)memory, (const float4*)evec, (const float4*)wvec,
        ws_ww, (float4*)out_mem);

    // B * (N/16)^2 tiles, 8 waves (tiles) per 256-thread block
    const int tiles  = B_ * (N_ / 16) * (N_ / 16);            // 262144
    dnc_link<<<tiles / 8, 256, 0, stream>>>(link, prec, ws_ww, out_link);
}